// ContextEncoding_19842748908208
// MI455X (gfx1250) — compile-verified
//
#include <hip/hip_runtime.h>
#include <math.h>

// ---------------------------------------------------------------------------
// CDNA5 (gfx1250) ContextEncoding: fp32 WMMA (V_WMMA_F32_16X16X4_F32) GEMMs
// ---------------------------------------------------------------------------

typedef __attribute__((ext_vector_type(2))) float v2f;
typedef __attribute__((ext_vector_type(8))) float v8f;

#define BB 4
#define CC 512
#define NN 16384           // H*W
#define KK 32
#define EPSV 1e-5f

__device__ __forceinline__ v8f wmma4(v2f a, v2f b, v8f c) {
  // D(16x16,f32) = A(16x4,f32) * B(4x16,f32) + C
  return __builtin_amdgcn_wmma_f32_16x16x4_f32(
      /*neg_a=*/false, a, /*neg_b=*/false, b,
      /*c_mod=*/(short)0, c, /*reuse_a=*/false, /*reuse_b=*/false);
}

// --------------------------- c2[k] = ||codeword_k||^2 -----------------------
__global__ void k_c2(const float* __restrict__ cw, float* __restrict__ c2) {
  int k = threadIdx.x;
  if (k < KK) {
    float s = 0.f;
    for (int c = 0; c < CC; ++c) { float v = cw[k * CC + c]; s += v * v; }
    c2[k] = s;
  }
}

// ------------- conv(1x1) GEMM + BN2 + ReLU  ->  P[b][c][n] ------------------
// grid (256 n-blocks, 2 m-blocks, 4 batches), block 256 (8 waves)
// wave tile: 32(M=out-ch) x 64(N=pixels): two A fragments amortize each B
// fragment (10 VMEM : 8 WMMA per K-step of 4 vs 9:4 with a 16-row tile).
__global__ void k_conv(const float* __restrict__ x, const float* __restrict__ Wc,
                       const float* __restrict__ g2, const float* __restrict__ b2,
                       const float* __restrict__ m2, const float* __restrict__ vv2,
                       float* __restrict__ P) {
  const int lane = threadIdx.x & 31;
  const int wave = threadIdx.x >> 5;
  const int half = lane >> 4;      // which 16-lane half
  const int l    = lane & 15;
  const int b    = blockIdx.z;
  const int m0   = blockIdx.y * 256 + wave * 32;
  const int n0   = blockIdx.x * 64;
  const float* X  = x + (size_t)b * CC * NN;
  float*       Pb = P + (size_t)b * CC * NN;

  v8f acc0[4] = {};            // rows m0   .. m0+15
  v8f acc1[4] = {};            // rows m0+16.. m0+31
  const int ar0 = m0 + l;
  const int ar1 = m0 + 16 + l;
  for (int c = 0; c < CC; c += 4) {
    const int ka = c + half * 2;  // lane-half selects K pair (A & B layouts)
    v2f a0 = *(const v2f*)(Wc + (size_t)ar0 * CC + ka);
    v2f a1 = *(const v2f*)(Wc + (size_t)ar1 * CC + ka);
    const float* X0 = X + (size_t)ka * NN + n0 + l;
    const float* X1 = X0 + NN;
#pragma unroll
    for (int j = 0; j < 4; ++j) {
      v2f bf; bf.x = X0[j * 16]; bf.y = X1[j * 16];
      acc0[j] = wmma4(a0, bf, acc0[j]);
      acc1[j] = wmma4(a1, bf, acc1[j]);
    }
  }
  // epilogue: BN2 (eval) + ReLU, store P[b][m][n]
#pragma unroll
  for (int t = 0; t < 2; ++t) {
    const v8f* acc = t ? acc1 : acc0;
#pragma unroll
    for (int r = 0; r < 8; ++r) {
      const int m = m0 + t * 16 + half * 8 + r;  // D: lanes>=16 hold M = r+8
      float rs = rsqrtf(vv2[m] + EPSV);
      float s  = g2[m] * rs;
      float tt = b2[m] - m2[m] * s;
      float* row = Pb + (size_t)m * NN + n0 + l;
#pragma unroll
      for (int j = 0; j < 4; ++j)
        row[j * 16] = fmaxf(acc[j][r] * s + tt, 0.f);
    }
  }
}

// ---- xc GEMM (codes x pixels) + in-flight ||z||^2 + in-register softmax ----
// grid (128 n-blocks, 4 batches), block 256 (8 waves); wave tile 32(K) x 16(N)
__global__ void k_xc_softmax(const float* __restrict__ P, const float* __restrict__ cw,
                             const float* __restrict__ scale, const float* __restrict__ c2,
                             float* __restrict__ assignbuf) {
  const int lane = threadIdx.x & 31;
  const int wave = threadIdx.x >> 5;
  const int half = lane >> 4;
  const int l    = lane & 15;
  const int b    = blockIdx.y;
  const int n0   = blockIdx.x * 128 + wave * 16;
  const int n    = n0 + l;
  const float* Pb = P + (size_t)b * CC * NN;

  v8f acc0 = {}, acc1 = {};
  float sq = 0.f;  // partial sum of z^2 for column n (this lane-half's K slice)
  for (int c = 0; c < CC; c += 4) {
    const int ka = c + half * 2;
    v2f a0 = *(const v2f*)(cw + (size_t)l * CC + ka);         // codes 0..15
    v2f a1 = *(const v2f*)(cw + (size_t)(16 + l) * CC + ka);  // codes 16..31
    v2f bf;
    bf.x = Pb[(size_t)ka * NN + n];
    bf.y = Pb[(size_t)(ka + 1) * NN + n];
    sq += bf.x * bf.x + bf.y * bf.y;
    acc0 = wmma4(a0, bf, acc0);
    acc1 = wmma4(a1, bf, acc1);
  }
  float x2 = sq + __shfl_xor(sq, 16, 32);  // join the two K-halves

  // scaled_l2 + softmax over 32 codes for pixel n (16 values per lane)
  float vals[16];
  float vmax = -3.402823466e38f;
#pragma unroll
  for (int r = 0; r < 8; ++r) {
    const int k0 = half * 8 + r;
    const int k1 = 16 + half * 8 + r;
    float v0 = scale[k0] * (x2 - 2.f * acc0[r] + c2[k0]);
    float v1 = scale[k1] * (x2 - 2.f * acc1[r] + c2[k1]);
    vals[r] = v0; vals[8 + r] = v1;
    vmax = fmaxf(vmax, fmaxf(v0, v1));
  }
  vmax = fmaxf(vmax, __shfl_xor(vmax, 16, 32));
  float ssum = 0.f;
#pragma unroll
  for (int i = 0; i < 16; ++i) { vals[i] = expf(vals[i] - vmax); ssum += vals[i]; }
  ssum += __shfl_xor(ssum, 16, 32);
  const float inv = 1.f / ssum;
#pragma unroll
  for (int r = 0; r < 8; ++r) {
    const int k0 = half * 8 + r;
    const int k1 = 16 + half * 8 + r;
    assignbuf[((size_t)(b * KK + k0)) * NN + n] = vals[r] * inv;
    assignbuf[((size_t)(b * KK + k1)) * NN + n] = vals[8 + r] * inv;
  }
}

// ---------- asum[b,k] = sum_n assign; agg init = -asum * codewords ----------
__global__ void k_asum_init(const float* __restrict__ assignbuf,
                            const float* __restrict__ cw, float* __restrict__ agg) {
  const int bk = blockIdx.x;       // b*32 + k
  const int k  = bk & (KK - 1);
  __shared__ float red[256];
  const float* row = assignbuf + (size_t)bk * NN;
  float s = 0.f;
  for (int n = threadIdx.x; n < NN; n += 256) s += row[n];
  red[threadIdx.x] = s;
  __syncthreads();
  for (int off = 128; off > 0; off >>= 1) {
    if (threadIdx.x < off) red[threadIdx.x] += red[threadIdx.x + off];
    __syncthreads();
  }
  const float asum = red[0];
  for (int c = threadIdx.x; c < CC; c += 256)
    agg[(size_t)bk * CC + c] = -asum * cw[k * CC + c];
}

// ------ agg GEMM: assign^T[32 x N] * z[N x 512], split-K over pixels --------
// grid (8 n-chunks, 4 c-groups, 4 batches), block 256; wave tile 32(K) x 16(C)
__global__ void k_agg(const float* __restrict__ P, const float* __restrict__ assignbuf,
                      float* __restrict__ agg) {
  const int lane = threadIdx.x & 31;
  const int wave = threadIdx.x >> 5;
  const int half = lane >> 4;
  const int l    = lane & 15;
  const int b    = blockIdx.z;
  const int c0   = (blockIdx.y * 8 + wave) * 16;
  const int nbase = blockIdx.x * 2048;
  const float* Pb = P + (size_t)b * CC * NN;
  const float* Ab = assignbuf + (size_t)b * KK * NN;

  v8f acc0 = {}, acc1 = {};
  for (int nn = 0; nn < 2048; nn += 4) {
    const int ka = nbase + nn + half * 2;  // K-dim = pixel index
    v2f a0 = *(const v2f*)(Ab + (size_t)l * NN + ka);
    v2f a1 = *(const v2f*)(Ab + (size_t)(16 + l) * NN + ka);
    v2f bf = *(const v2f*)(Pb + (size_t)(c0 + l) * NN + ka);  // z[n][c]=P[c][n]
    acc0 = wmma4(a0, bf, acc0);
    acc1 = wmma4(a1, bf, acc1);
  }
#pragma unroll
  for (int r = 0; r < 8; ++r) {
    const int k0 = half * 8 + r;
    const int k1 = 16 + half * 8 + r;
    atomicAdd(&agg[((size_t)(b * KK + k0)) * CC + c0 + l], acc0[r]);
    atomicAdd(&agg[((size_t)(b * KK + k1)) * CC + c0 + l], acc1[r]);
  }
}

// -------------- BN1 + ReLU + mean over codes  -> feat[b][c] ----------------
__global__ void k_feat(const float* __restrict__ agg, const float* __restrict__ g1,
                       const float* __restrict__ b1, const float* __restrict__ m1,
                       const float* __restrict__ v1, float* __restrict__ feat) {
  const int idx = blockIdx.x * 256 + threadIdx.x;  // 0..2047
  const int b = idx >> 9, c = idx & (CC - 1);
  float accv = 0.f;
#pragma unroll
  for (int k = 0; k < KK; ++k) {
    float s = g1[k] * rsqrtf(v1[k] + EPSV);
    float t = b1[k] - m1[k] * s;
    float v = agg[((size_t)(b * KK + k)) * CC + c] * s + t;
    accv += fmaxf(v, 0.f);
  }
  feat[idx] = accv * (1.f / (float)KK);
}

// -------------------- gamma = sigmoid(feat @ fc_w^T + b) -------------------
__global__ void k_fc(const float* __restrict__ feat, const float* __restrict__ fcw,
                     const float* __restrict__ fcb, float* __restrict__ gamma) {
  const int idx = blockIdx.x * 256 + threadIdx.x;  // 0..2047
  const int b = idx >> 9, o = idx & (CC - 1);
  const float* f = feat + (size_t)b * CC;
  const float* w = fcw + (size_t)o * CC;
  float accv = fcb[o];
  for (int c = 0; c < CC; ++c) accv += f[c] * w[c];
  gamma[idx] = 1.f / (1.f + expf(-accv));
}

// ----------------- out = relu(x * (1 + gamma)), float4 pass ----------------
__global__ void k_out(const float* __restrict__ x, const float* __restrict__ gamma,
                      float* __restrict__ out) {
  const size_t i4 = (size_t)blockIdx.x * 256 + threadIdx.x;  // float4 index
  const size_t i = i4 * 4;
  const int bc = (int)(i >> 14);  // i / NN == b*512 + c
  const float g = 1.f + gamma[bc];
  const float4 xv = ((const float4*)x)[i4];
  float4 o;
  o.x = fmaxf(xv.x * g, 0.f);
  o.y = fmaxf(xv.y * g, 0.f);
  o.z = fmaxf(xv.z * g, 0.f);
  o.w = fmaxf(xv.w * g, 0.f);
  ((float4*)out)[i4] = o;
}

// ---------------------------------------------------------------------------
extern "C" void kernel_launch(void* const* d_in, const int* in_sizes, int n_in,
                              void* d_out, int out_size, void* d_ws, size_t ws_size,
                              hipStream_t stream) {
  (void)in_sizes; (void)n_in; (void)out_size; (void)ws_size;
  const float* x      = (const float*)d_in[0];
  const float* conv_w = (const float*)d_in[1];
  const float* bn2_g  = (const float*)d_in[2];
  const float* bn2_b  = (const float*)d_in[3];
  const float* bn2_m  = (const float*)d_in[4];
  const float* bn2_v  = (const float*)d_in[5];
  const float* cw     = (const float*)d_in[6];
  const float* scale  = (const float*)d_in[7];
  const float* bn1_g  = (const float*)d_in[8];
  const float* bn1_b  = (const float*)d_in[9];
  const float* bn1_m  = (const float*)d_in[10];
  const float* bn1_v  = (const float*)d_in[11];
  const float* fc_w   = (const float*)d_in[12];
  const float* fc_b   = (const float*)d_in[13];

  float* feat = (float*)d_out;               // [B,C] = 2048 floats
  float* outp = (float*)d_out + BB * CC;     // [B,C,H,W]

  // workspace layout (floats): P | assign | agg | c2 | gamma  (~143 MB)
  float* P         = (float*)d_ws;
  float* assignbuf = P + (size_t)BB * CC * NN;
  float* agg       = assignbuf + (size_t)BB * KK * NN;
  float* c2        = agg + (size_t)BB * KK * CC;
  float* gamma     = c2 + KK;

  k_c2<<<1, 32, 0, stream>>>(cw, c2);
  k_conv<<<dim3(NN / 64, CC / 256, BB), 256, 0, stream>>>(
      x, conv_w, bn2_g, bn2_b, bn2_m, bn2_v, P);
  k_xc_softmax<<<dim3(NN / 128, BB), 256, 0, stream>>>(P, cw, scale, c2, assignbuf);
  k_asum_init<<<BB * KK, 256, 0, stream>>>(assignbuf, cw, agg);
  k_agg<<<dim3(NN / 2048, 4, BB), 256, 0, stream>>>(P, assignbuf, agg);
  k_feat<<<(BB * CC) / 256, 256, 0, stream>>>(agg, bn1_g, bn1_b, bn1_m, bn1_v, feat);
  k_fc<<<(BB * CC) / 256, 256, 0, stream>>>(feat, fc_w, fc_b, gamma);
  k_out<<<(BB * CC * NN / 4) / 256, 256, 0, stream>>>(x, gamma, outp);
}